// VoxelProjection_31258771980988
// MI455X (gfx1250) — compile-verified
//
#include <hip/hip_runtime.h>

// Project2Dto3D gather: out[d,c,hb,wb] = sum_n input[n,c,v,u] * valid*density
// Memory-bound gather-reduce. No matmul structure -> no WMMA; instead use the
// CDNA5 async-to-LDS pipeline (ASYNCcnt) for the scattered L2 gathers and
// non-temporal stores so the 90.8 MB feature map stays resident in the 192 MB L2.

namespace {
constexpr int kNCam = 4;
constexpr int kC    = 336;
constexpr int kH    = 96;
constexpr int kW    = 176;
constexpr int kD    = 6;
constexpr int kHB   = 240;
constexpr int kWB   = 120;

constexpr int  kHW     = kH * kW;          // 16896 (per-channel plane)
constexpr long kCHW    = (long)kC * kHW;   // per-camera stride
constexpr int  kHBWB   = kHB * kWB;        // 28800 (voxel plane)
constexpr int  kNVox   = kD * kHBWB;       // 172800 = 675 * 256 exactly
constexpr int  kTPB    = 256;              // 8 wave32s per block
constexpr int  kDepth  = 4;                // async pipeline depth (channels in flight)
constexpr unsigned kBufBytes = kTPB * 16u; // one float4 slot per thread per stage
constexpr int  kMain   = 332;              // 83 * 4 main-loop channels; 332..335 peeled
}

// Wait until ASYNCcnt <= n (async loads complete in order).
#define WAIT_ASYNC(n) asm volatile("s_wait_asynccnt %0" :: "i"(n) : "memory")

// GVS async gather: copies 4 bytes from global address cam_base + goff_bytes to the LDS address lds_bytes
__device__ __forceinline__ void issue_async_gather(const float* cam_base,
                                                   unsigned goff_bytes,
                                                   unsigned lds_bytes)
{
  asm volatile("global_load_async_to_lds_b32 %0, %1, %2"
               :: "v"(lds_bytes), "v"(goff_bytes), "s"(cam_base)
               : "memory");
}

__global__ void __launch_bounds__(kTPB)
voxproj_gather_kernel(const float* __restrict__ feat,
                      const int*   __restrict__ pu,
                      const int*   __restrict__ pv,
                      const int*   __restrict__ pvalid,
                      const float* __restrict__ pdens,
                      float*       __restrict__ out)
{
  __shared__ float4 stage[kDepth * kTPB];   // 16 KB of the 320 KB WGP pool

  const int tid = threadIdx.x;
  const int vox = blockIdx.x * kTPB + tid;  // grid sized exactly: no guard
  const int d   = vox / kHBWB;
  const int p   = vox - d * kHBWB;          // lane-consecutive along wb

  unsigned     goff[kNCam];
  float        w[kNCam];
  const float* base[kNCam];
#pragma unroll
  for (int n = 0; n < kNCam; ++n) {
    const int idx = n * kNVox + vox;        // coalesced projection-table reads
    const int u = pu[idx];
    const int v = pv[idx];
    goff[n] = (unsigned)((v * kW + u) * 4); // byte offset of (v,u) in a channel plane
    // Branchless, matches reference: w = float(valid) * density.
    w[n]    = (float)pvalid[idx] * pdens[idx];
    base[n] = feat + (size_t)n * (size_t)kCHW;
  }

  // addrspace(3) offset = low 32 bits of the flat shared address (ISA: LDS
  // aperture keeps the LDS byte offset in addr[31:0]).
  const unsigned lds0 = (unsigned)(size_t)(&stage[0]) + (unsigned)tid * 16u;

  // Prologue: fill kDepth-1 pipeline stages (channels 0..2).
#pragma unroll
  for (int c = 0; c < kDepth - 1; ++c) {
    const unsigned cb = (unsigned)c * (unsigned)(kHW * 4);
    const unsigned lb = lds0 + (unsigned)(c & (kDepth - 1)) * kBufBytes;
#pragma unroll
    for (int n = 0; n < kNCam; ++n)
      issue_async_gather(base[n], goff[n] + cb, lb + (unsigned)n * 4u);
  }

  float* __restrict__ outp = out + (size_t)d * kC * kHBWB + p;

  // One ds_load_b128 pulls the 4 gathered camera values for this channel;
  // the 4-term weighted reduction feeds a non-temporal store (write-once data,
  // keep it out of L2 so the feature map stays resident).
#define CONSUME(cc) do {                                                     \
    const float4 f = stage[((cc) & (kDepth - 1)) * kTPB + tid];              \
    const float acc = f.x * w[0] + f.y * w[1] + f.z * w[2] + f.w * w[3];     \
    __builtin_nontemporal_store(acc, outp + (size_t)(cc) * kHBWB);           \
  } while (0)

#define STEP(cc) do {                                                        \
    WAIT_ASYNC(8); /* batch cc retired (in-order completion) */              \
    CONSUME(cc);                                                             \
    const int      cn_ = (cc) + (kDepth - 1);                                \
    const unsigned cb_ = (unsigned)cn_ * (unsigned)(kHW * 4);                \
    const unsigned lb_ = lds0 + (unsigned)(cn_ & (kDepth - 1)) * kBufBytes;  \
    _Pragma("unroll")                                                        \
    for (int n_ = 0; n_ < kNCam; ++n_)                                       \
      issue_async_gather(base[n_], goff[n_] + cb_, lb_ + (unsigned)n_ * 4u); \
  } while (0)

  // Steady state: up to 3 batches (12 async loads) in flight per wave.
  // wait <= 8 retires the oldest batch, consume it, reuse channel c-1's buffer for
  // channel c+3 (drained: its ds_load was consumed in the previous step).
  // Outer loop kept as a loop (83 iters); inner 4 steps fully unrolled so
  // each copy's (c & 3) stage address folds to an immediate.
#pragma clang loop unroll(disable)
  for (int c0 = 0; c0 < kMain; c0 += kDepth) {
#pragma unroll
    for (int j = 0; j < kDepth; ++j)
      STEP(c0 + j);
  }

  // Peeled final issue step (c = 332 issues batch 335), then drain.
  STEP(kMain);                        // wait 8, consume 332, issue 335
  WAIT_ASYNC(8);  CONSUME(kC - 3);    // 333
  WAIT_ASYNC(4);  CONSUME(kC - 2);    // 334
  WAIT_ASYNC(0);  CONSUME(kC - 1);    // 335
#undef STEP
#undef CONSUME
}

extern "C" void kernel_launch(void* const* d_in, const int* in_sizes, int n_in,
                              void* d_out, int out_size, void* d_ws, size_t ws_size,
                              hipStream_t stream)
{
  (void)in_sizes; (void)n_in; (void)out_size; (void)d_ws; (void)ws_size;

  const float* feat   = (const float*)d_in[0];
  const int*   pu     = (const int*)  d_in[1];
  const int*   pv     = (const int*)  d_in[2];
  const int*   pvalid = (const int*)  d_in[3];
  const float* pdens  = (const float*)d_in[4];
  float*       out    = (float*)d_out;

  const dim3 grid(kNVox / kTPB);  // 675 blocks, exact
  const dim3 block(kTPB);
  voxproj_gather_kernel<<<grid, block, 0, stream>>>(feat, pu, pv, pvalid, pdens, out);
}